// LocalFeatureAggregation_86852828660097
// MI455X (gfx1250) — compile-verified
//
#include <hip/hip_runtime.h>
#include <hip/hip_bf16.h>
#include <math.h>

// Problem constants (match reference)
#define NB 2
#define NPTS 32768
#define KNN 16
#define DIN 8
#define D2C 16
#define DOUTC 32

// Flip to 0 if the async-to-LDS inline asm fails to assemble.
#define USE_ASYNC_LDS 1

typedef __attribute__((ext_vector_type(2))) float v2f;
typedef __attribute__((ext_vector_type(8))) float v8f;

// ---------------------------------------------------------------------------
// Workspace layout (bytes)
// ---------------------------------------------------------------------------
#define OFF_IDX      ((size_t)0)            //  4 MB  int   (B,N,K)
#define OFF_DIST     ((size_t)(4  << 20))   //  4 MB  float (B,N,K)
#define OFF_X1       ((size_t)(8  << 20))   //  4 MB  float (B,16,N)  leaky(mlp1)
#define OFF_SCPRE    ((size_t)(12 << 20))   // 16 MB  float (B,64,N)  shortcut pre-BN
#define OFF_P1       ((size_t)(28 << 20))   //  4 MB  float (B,16,N)  pool1 pre-BN
#define OFF_P2       ((size_t)(32 << 20))   //  8 MB  float (B,32,N)  pool2 pre-BN
#define OFF_PART_LSE ((size_t)(40 << 20))   // 512 KB partials 4096 x 16 x 2
#define OFF_PART_SC  ((size_t)(41 << 20))   //  8 KB  partials 16 x 64 x 2
#define OFF_PART_P   (OFF_PART_SC + 65536)  //  4 KB  partials 16 x 32 x 2
#define OFF_ST_SC    (OFF_PART_P + 65536)   // per-channel (scale,shift) pairs
#define OFF_ST_LSE1  (OFF_ST_SC   + 1024)
#define OFF_ST_LSE2  (OFF_ST_LSE1 + 1024)
#define OFF_ST_P1    (OFF_ST_LSE2 + 1024)
#define OFF_ST_P2    (OFF_ST_P1   + 1024)

// ---------------------------------------------------------------------------
// 1) KNN via WMMA:  A = candidates (px,py,pz,p^2), B = queries (-2qx,-2qy,-2qz,1)
//    D[p][q] = p^2 - 2 q.p  ->  d2 = D + q^2 (lane-constant). The p^2 term is
//    folded into the 4th K slot, so selection needs NO LDS traffic at all.
//    Lane l owns query column l&15; lanes l and l+16 each scan half the
//    candidate rows (rbase 0 / 8) and keep private sorted top-16 lists that are
//    merged through LDS at the end.
// ---------------------------------------------------------------------------
#define CCH 128   // candidate chunk staged in LDS (shared by 8 waves)

__global__ __launch_bounds__(256) void knn_kernel(const float* __restrict__ coords,
                                                  int* __restrict__ oidx,
                                                  float* __restrict__ odist) {
  __shared__ float sx[CCH], sy[CCH], sz[CCH], sp2[CCH];
#if USE_ASYNC_LDS
  __shared__ float sc3[CCH * 3];           // raw AoS staging target for async DMA
#endif
  __shared__ float md[8][32][16];          // per-wave merge scratch (dists)
  __shared__ int   mi[8][32][16];          // per-wave merge scratch (indices)

  const int tid  = threadIdx.x;
  const int wave = tid >> 5;
  const int lane = tid & 31;
  const int blocksPerBatch = NPTS / 128;   // 8 waves * 16 queries
  const int b     = blockIdx.x / blocksPerBatch;
  const int qblk  = blockIdx.x % blocksPerBatch;
  const int qbase = qblk * 128 + wave * 16;
  const float* cb = coords + (size_t)b * NPTS * 3;

  // Query for this lane's D column.
  const int q = qbase + (lane & 15);
  const float qx = cb[q * 3 + 0];
  const float qy = cb[q * 3 + 1];
  const float qz = cb[q * 3 + 2];
  const float q2 = qx * qx + qy * qy + qz * qz;

  // B operand (4x16 queries), resident in registers all kernel:
  // VGPR0: lanes0-15 K=0 (-2qx), lanes16-31 K=2 (-2qz)
  // VGPR1: lanes0-15 K=1 (-2qy), lanes16-31 K=3 (1.0)
  v2f bq;
  bq.x = (lane < 16) ? (-2.0f * qx) : (-2.0f * qz);
  bq.y = (lane < 16) ? (-2.0f * qy) : 1.0f;

  const int rbase = (lane < 16) ? 0 : 8;   // candidate-row half owned by lane

  float bd[KNN];
  int   bi[KNN];
#pragma unroll
  for (int j = 0; j < KNN; ++j) { bd[j] = 3.4e38f; bi[j] = 0; }

  for (int c0 = 0; c0 < NPTS; c0 += CCH) {
    __syncthreads();
#if USE_ASYNC_LDS
    {
      // Async DMA the 384-dword coord chunk straight into LDS (ASYNCcnt path).
      const float* gsrc = cb + (size_t)c0 * 3;
      {
        const unsigned l0 = (unsigned)(size_t)&sc3[tid];
        const float*   g0 = gsrc + tid;
        asm volatile("global_load_async_to_lds_b32 %0, %1, off"
                     :: "v"(l0), "v"(g0) : "memory");
      }
      if (tid < CCH) {
        const unsigned l1 = (unsigned)(size_t)&sc3[256 + tid];
        const float*   g1 = gsrc + 256 + tid;
        asm volatile("global_load_async_to_lds_b32 %0, %1, off"
                     :: "v"(l1), "v"(g1) : "memory");
      }
      asm volatile("s_wait_asynccnt 0" ::: "memory");
      __syncthreads();
      if (tid < CCH) {
        const float px = sc3[tid * 3 + 0];
        const float py = sc3[tid * 3 + 1];
        const float pz = sc3[tid * 3 + 2];
        sx[tid] = px; sy[tid] = py; sz[tid] = pz;
        sp2[tid] = px * px + py * py + pz * pz;
      }
    }
#else
    if (tid < CCH) {
      const float px = cb[(c0 + tid) * 3 + 0];
      const float py = cb[(c0 + tid) * 3 + 1];
      const float pz = cb[(c0 + tid) * 3 + 2];
      sx[tid] = px; sy[tid] = py; sz[tid] = pz;
      sp2[tid] = px * px + py * py + pz * pz;
      if (c0 + CCH < NPTS)
        __builtin_prefetch(&cb[(c0 + CCH + tid) * 3], 0, 1);
    }
#endif
    __syncthreads();

    for (int sub = 0; sub < CCH / 16; ++sub) {
      // A operand (16 candidates x 4):
      // lanes0-15: (px,py) of cand cc; lanes16-31: (pz,p2) of cand cc
      const int cc = sub * 16 + (lane & 15);
      v2f a;
      a.x = (lane < 16) ? sx[cc] : sz[cc];
      a.y = (lane < 16) ? sy[cc] : sp2[cc];

      v8f d = {};
      d = __builtin_amdgcn_wmma_f32_16x16x4_f32(false, a, false, bq,
                                                (short)0, d, false, false);

      const int cibase = c0 + sub * 16 + rbase;
#pragma unroll
      for (int j = 0; j < 8; ++j) {
        const float d2 = d[j] + q2;
        const int   ci = cibase + j;
        if (d2 < bd[KNN - 1]) {
          bd[KNN - 1] = d2; bi[KNN - 1] = ci;
#pragma unroll
          for (int t = KNN - 1; t > 0; --t) {
            if (bd[t] < bd[t - 1]) {
              const float td = bd[t]; bd[t] = bd[t - 1]; bd[t - 1] = td;
              const int   ti = bi[t]; bi[t] = bi[t - 1]; bi[t - 1] = ti;
            }
          }
        }
      }
    }
  }

  // Merge the two half-lists (lanes l and l+16) per query through LDS.
  __syncthreads();
#pragma unroll
  for (int j = 0; j < KNN; ++j) { md[wave][lane][j] = bd[j]; mi[wave][lane][j] = bi[j]; }
  __syncthreads();

  if (lane < 16) {
    const float* Ad = md[wave][lane];
    const float* Bd = md[wave][lane + 16];
    const int*   Ai = mi[wave][lane];
    const int*   Bi = mi[wave][lane + 16];
    int ia = 0, ib = 0;
    const size_t base = ((size_t)b * NPTS + (qbase + lane)) * KNN;
#pragma unroll
    for (int t = 0; t < KNN; ++t) {
      const float da = Ad[ia], db = Bd[ib];
      const int   xa = Ai[ia], xb = Bi[ib];
      const bool takeA = (da < db) || (da == db && xa < xb);  // top_k tie: low idx
      const float dsel = takeA ? da : db;
      const int   isel = takeA ? xa : xb;
      ia += takeA ? 1 : 0;
      ib += takeA ? 0 : 1;
      oidx[base + t] = isel;
      const float x = dsel > 0.0f ? dsel : 0.0f;
      odist[base + t] = x > 0.0f ? sqrtf(x) : 0.0f;
    }
  }
}

// ---------------------------------------------------------------------------
// 2) mlp1 (leaky 0.2) + shortcut pre-activation
// ---------------------------------------------------------------------------
__global__ __launch_bounds__(256) void mlp1_sc_kernel(const float* __restrict__ feat,
                                                      const float* __restrict__ w1,
                                                      const float* __restrict__ b1,
                                                      const float* __restrict__ ws_,
                                                      const float* __restrict__ bs,
                                                      float* __restrict__ x1,
                                                      float* __restrict__ scpre) {
  const int t = blockIdx.x * 256 + threadIdx.x;
  const int b = t / NPTS, n = t % NPTS;
  float f[DIN];
#pragma unroll
  for (int c = 0; c < DIN; ++c) f[c] = feat[(size_t)(b * DIN + c) * NPTS + n];
#pragma unroll
  for (int o = 0; o < D2C; ++o) {
    float v = b1[o];
#pragma unroll
    for (int c = 0; c < DIN; ++c) v = fmaf(w1[o * DIN + c], f[c], v);
    v = v >= 0.0f ? v : 0.2f * v;
    x1[(size_t)(b * D2C + o) * NPTS + n] = v;
  }
#pragma unroll
  for (int j = 0; j < 2 * DOUTC; ++j) {
    float v = bs[j];
#pragma unroll
    for (int c = 0; c < DIN; ++c) v = fmaf(ws_[j * DIN + c], f[c], v);
    scpre[(size_t)(b * 2 * DOUTC + j) * NPTS + n] = v;
  }
}

// ---------------------------------------------------------------------------
// 3) LSE BN statistics: recompute 16-ch encoder per (b,n,k), deterministic
// ---------------------------------------------------------------------------
__global__ __launch_bounds__(256) void lse_stats_kernel(const float* __restrict__ coords,
                                                        const int* __restrict__ idx,
                                                        const float* __restrict__ dist,
                                                        const float* __restrict__ lw,
                                                        const float* __restrict__ lb,
                                                        float* __restrict__ partial) {
  __shared__ float sm[256 * 16];
  const int tid = threadIdx.x;
  const int t = blockIdx.x * 256 + tid;
  const int b = t / (NPTS * KNN);
  const int r = t % (NPTS * KNN);
  const int n = r / KNN;
  const int id = idx[t];
  const float dd = dist[t];
  const float* cb = coords + (size_t)b * NPTS * 3;
  const float cx = cb[n * 3], cy = cb[n * 3 + 1], cz = cb[n * 3 + 2];
  const float nx = cb[id * 3], ny = cb[id * 3 + 1], nz = cb[id * 3 + 2];
  const float rel[10] = {cx, cy, cz, nx, ny, nz, cx - nx, cy - ny, cz - nz, dd};

  float enc[16];
#pragma unroll
  for (int o = 0; o < 16; ++o) {
    float e = lb[o];
#pragma unroll
    for (int c = 0; c < 10; ++c) e = fmaf(lw[o * 10 + c], rel[c], e);
    enc[o] = e;
  }
#pragma unroll
  for (int o = 0; o < 16; ++o) sm[tid * 16 + o] = enc[o];
  __syncthreads();
  for (int s = 128; s > 0; s >>= 1) {
    if (tid < s) {
#pragma unroll
      for (int o = 0; o < 16; ++o) sm[tid * 16 + o] += sm[(tid + s) * 16 + o];
    }
    __syncthreads();
  }
  if (tid < 16) partial[(size_t)blockIdx.x * 32 + tid * 2] = sm[tid];
  __syncthreads();
#pragma unroll
  for (int o = 0; o < 16; ++o) sm[tid * 16 + o] = enc[o] * enc[o];
  __syncthreads();
  for (int s = 128; s > 0; s >>= 1) {
    if (tid < s) {
#pragma unroll
      for (int o = 0; o < 16; ++o) sm[tid * 16 + o] += sm[(tid + s) * 16 + o];
    }
    __syncthreads();
  }
  if (tid < 16) partial[(size_t)blockIdx.x * 32 + tid * 2 + 1] = sm[tid];
}

// ---------------------------------------------------------------------------
// 4) Generic BN statistics over a materialized (B,C,N) array, 16 slabs
// ---------------------------------------------------------------------------
__global__ __launch_bounds__(256) void stats_partial_kernel(const float* __restrict__ x,
                                                            int C, float* __restrict__ partial) {
  __shared__ float sm[256];
  const int c = blockIdx.x, slab = blockIdx.y, tid = threadIdx.x;
  const int per = (NB * NPTS) / 16;
  float s = 0.0f, q = 0.0f;
  for (int i = tid; i < per; i += 256) {
    const int g = slab * per + i;
    const int b = g / NPTS, n = g % NPTS;
    const float v = x[(size_t)(b * C + c) * NPTS + n];
    s += v; q = fmaf(v, v, q);
  }
  sm[tid] = s; __syncthreads();
  for (int st = 128; st > 0; st >>= 1) { if (tid < st) sm[tid] += sm[tid + st]; __syncthreads(); }
  if (tid == 0) partial[(size_t)(slab * C + c) * 2] = sm[0];
  __syncthreads();
  sm[tid] = q; __syncthreads();
  for (int st = 128; st > 0; st >>= 1) { if (tid < st) sm[tid] += sm[tid + st]; __syncthreads(); }
  if (tid == 0) partial[(size_t)(slab * C + c) * 2 + 1] = sm[0];
}

// Fold (gamma, beta, mean, var) -> per-channel (scale, shift)
__global__ void bn_final_kernel(const float* __restrict__ partial, int nPart, int C,
                                const float* __restrict__ g, const float* __restrict__ be,
                                float cnt, float* __restrict__ stats) {
  const int c = threadIdx.x;
  if (c >= C) return;
  float s = 0.0f, q = 0.0f;
  for (int p = 0; p < nPart; ++p) {
    s += partial[(size_t)(p * C + c) * 2];
    q += partial[(size_t)(p * C + c) * 2 + 1];
  }
  const float m = s / cnt;
  const float v = q / cnt - m * m;
  const float inv = rsqrtf(v + 1e-6f);
  const float sc = g[c] * inv;
  stats[c * 2] = sc;
  stats[c * 2 + 1] = be[c] - m * sc;
}

// ---------------------------------------------------------------------------
// 5) Attentive pooling: one wave per point
// ---------------------------------------------------------------------------
__global__ __launch_bounds__(256) void pool_kernel(const float* __restrict__ coords,
                                                   const int* __restrict__ idx,
                                                   const float* __restrict__ dist,
                                                   const float* __restrict__ lw,
                                                   const float* __restrict__ lb,
                                                   const float* __restrict__ lst,
                                                   const float* __restrict__ feats,
                                                   const float* __restrict__ fst,
                                                   int featBN,
                                                   const float* __restrict__ sW,
                                                   const float* __restrict__ mW,
                                                   const float* __restrict__ mb,
                                                   int COUT, float* __restrict__ out) {
  __shared__ float xb[8][32 * 17];
  __shared__ float fl[8][32];
  const int tid = threadIdx.x, w = tid >> 5, lane = tid & 31;
  const int p = blockIdx.x * 8 + w;
  const int b = p / NPTS, n = p % NPTS;
  float* X = xb[w];

  if (lane < 16) {                       // BN+relu'd encoder channels 0..15
    const int k = lane;
    const size_t t = ((size_t)(b * NPTS + n)) * KNN + k;
    const int id = idx[t];
    const float dd = dist[t];
    const float* cb = coords + (size_t)b * NPTS * 3;
    const float cx = cb[n * 3], cy = cb[n * 3 + 1], cz = cb[n * 3 + 2];
    const float nx = cb[id * 3], ny = cb[id * 3 + 1], nz = cb[id * 3 + 2];
    const float rel[10] = {cx, cy, cz, nx, ny, nz, cx - nx, cy - ny, cz - nz, dd};
#pragma unroll
    for (int o = 0; o < 16; ++o) {
      float e = lb[o];
#pragma unroll
      for (int c = 0; c < 10; ++c) e = fmaf(lw[o * 10 + c], rel[c], e);
      e = e * lst[o * 2] + lst[o * 2 + 1];
      e = e > 0.0f ? e : 0.0f;
      X[o * 17 + k] = e;
    }
  } else {                               // broadcast feats into channels 16..31
    const int c = lane - 16;
    float f = feats[(size_t)(b * 16 + c) * NPTS + n];
    if (featBN) { f = f * fst[c * 2] + fst[c * 2 + 1]; f = f > 0.0f ? f : 0.0f; }
#pragma unroll
    for (int k = 0; k < KNN; ++k) X[(16 + c) * 17 + k] = f;
  }
  __syncthreads();

  float s[KNN];
  float mx = -3.4e38f;
#pragma unroll
  for (int k = 0; k < KNN; ++k) {
    float a = 0.0f;
#pragma unroll
    for (int c = 0; c < 32; ++c) a = fmaf(sW[lane * 32 + c], X[c * 17 + k], a);
    s[k] = a;
    mx = a > mx ? a : mx;
  }
  float se = 0.0f;
#pragma unroll
  for (int k = 0; k < KNN; ++k) { s[k] = __expf(s[k] - mx); se += s[k]; }
  const float inv = 1.0f / se;
  float ft = 0.0f;
#pragma unroll
  for (int k = 0; k < KNN; ++k) ft = fmaf(s[k] * inv, X[lane * 17 + k], ft);
  fl[w][lane] = ft;
  __syncthreads();

  if (lane < COUT) {
    float a = mb[lane];
#pragma unroll
    for (int c = 0; c < 32; ++c) a = fmaf(mW[lane * 32 + c], fl[w][c], a);
    out[(size_t)(b * COUT + lane) * NPTS + n] = a;
  }
}

// ---------------------------------------------------------------------------
// 6) Final: BN+relu(pool2) -> mlp2 + BN(shortcut) -> leaky 0.01
// ---------------------------------------------------------------------------
__global__ __launch_bounds__(256) void final_kernel(const float* __restrict__ p2,
                                                    const float* __restrict__ st2,
                                                    const float* __restrict__ scpre,
                                                    const float* __restrict__ sts,
                                                    const float* __restrict__ w2,
                                                    const float* __restrict__ b2,
                                                    float* __restrict__ out) {
  const int t = blockIdx.x * 256 + threadIdx.x;
  const int b = t / NPTS, n = t % NPTS;
  float x2[DOUTC];
#pragma unroll
  for (int o = 0; o < DOUTC; ++o) {
    float v = p2[(size_t)(b * DOUTC + o) * NPTS + n];
    v = v * st2[o * 2] + st2[o * 2 + 1];
    x2[o] = v > 0.0f ? v : 0.0f;
  }
#pragma unroll
  for (int j = 0; j < 2 * DOUTC; ++j) {
    float acc = b2[j];
#pragma unroll
    for (int o = 0; o < DOUTC; ++o) acc = fmaf(w2[j * DOUTC + o], x2[o], acc);
    const float sc = scpre[(size_t)(b * 2 * DOUTC + j) * NPTS + n] * sts[j * 2] + sts[j * 2 + 1];
    acc += sc;
    out[(size_t)(b * 2 * DOUTC + j) * NPTS + n] = acc >= 0.0f ? acc : 0.01f * acc;
  }
}

// ---------------------------------------------------------------------------
extern "C" void kernel_launch(void* const* d_in, const int* in_sizes, int n_in,
                              void* d_out, int out_size, void* d_ws, size_t ws_size,
                              hipStream_t stream) {
  const float* coords   = (const float*)d_in[0];
  const float* features = (const float*)d_in[1];
  const float* mlp1_W   = (const float*)d_in[2];
  const float* mlp1_b   = (const float*)d_in[3];
  const float* lse1_W   = (const float*)d_in[4];
  const float* lse1_b   = (const float*)d_in[5];
  const float* lse1_g   = (const float*)d_in[6];
  const float* lse1_be  = (const float*)d_in[7];
  const float* pool1_sW = (const float*)d_in[8];
  const float* pool1_W  = (const float*)d_in[9];
  const float* pool1_b  = (const float*)d_in[10];
  const float* pool1_g  = (const float*)d_in[11];
  const float* pool1_be = (const float*)d_in[12];
  const float* lse2_W   = (const float*)d_in[13];
  const float* lse2_b   = (const float*)d_in[14];
  const float* lse2_g   = (const float*)d_in[15];
  const float* lse2_be  = (const float*)d_in[16];
  const float* pool2_sW = (const float*)d_in[17];
  const float* pool2_W  = (const float*)d_in[18];
  const float* pool2_b  = (const float*)d_in[19];
  const float* pool2_g  = (const float*)d_in[20];
  const float* pool2_be = (const float*)d_in[21];
  const float* mlp2_W   = (const float*)d_in[22];
  const float* mlp2_b   = (const float*)d_in[23];
  const float* sc_W     = (const float*)d_in[24];
  const float* sc_b     = (const float*)d_in[25];
  const float* sc_g     = (const float*)d_in[26];
  const float* sc_be    = (const float*)d_in[27];

  char* ws = (char*)d_ws;
  int*   idx      = (int*)  (ws + OFF_IDX);
  float* dist     = (float*)(ws + OFF_DIST);
  float* x1       = (float*)(ws + OFF_X1);
  float* scpre    = (float*)(ws + OFF_SCPRE);
  float* p1       = (float*)(ws + OFF_P1);
  float* p2       = (float*)(ws + OFF_P2);
  float* partLse  = (float*)(ws + OFF_PART_LSE);
  float* partSc   = (float*)(ws + OFF_PART_SC);
  float* partP    = (float*)(ws + OFF_PART_P);
  float* stSc     = (float*)(ws + OFF_ST_SC);
  float* stLse1   = (float*)(ws + OFF_ST_LSE1);
  float* stLse2   = (float*)(ws + OFF_ST_LSE2);
  float* stP1     = (float*)(ws + OFF_ST_P1);
  float* stP2     = (float*)(ws + OFF_ST_P2);
  float* out      = (float*)d_out;

  const float cntBN  = (float)(NB * NPTS);
  const float cntLse = (float)(NB * NPTS * KNN);
  const int nLseBlk  = (NB * NPTS * KNN) / 256;   // 4096

  // KNN via WMMA tiles (+ async global->LDS chunk staging)
  knn_kernel<<<dim3(NB * (NPTS / 128)), 256, 0, stream>>>(coords, idx, dist);

  // mlp1 + shortcut pre-act
  mlp1_sc_kernel<<<(NB * NPTS) / 256, 256, 0, stream>>>(features, mlp1_W, mlp1_b,
                                                        sc_W, sc_b, x1, scpre);
  // shortcut BN stats
  stats_partial_kernel<<<dim3(2 * DOUTC, 16), 256, 0, stream>>>(scpre, 2 * DOUTC, partSc);
  bn_final_kernel<<<1, 2 * DOUTC, 0, stream>>>(partSc, 16, 2 * DOUTC, sc_g, sc_be, cntBN, stSc);

  // lse1 stats + pool1
  lse_stats_kernel<<<nLseBlk, 256, 0, stream>>>(coords, idx, dist, lse1_W, lse1_b, partLse);
  bn_final_kernel<<<1, D2C, 0, stream>>>(partLse, nLseBlk, D2C, lse1_g, lse1_be, cntLse, stLse1);
  pool_kernel<<<(NB * NPTS) / 8, 256, 0, stream>>>(coords, idx, dist, lse1_W, lse1_b, stLse1,
                                                   x1, (const float*)nullptr, 0,
                                                   pool1_sW, pool1_W, pool1_b, D2C, p1);
  stats_partial_kernel<<<dim3(D2C, 16), 256, 0, stream>>>(p1, D2C, partP);
  bn_final_kernel<<<1, D2C, 0, stream>>>(partP, 16, D2C, pool1_g, pool1_be, cntBN, stP1);

  // lse2 stats + pool2
  lse_stats_kernel<<<nLseBlk, 256, 0, stream>>>(coords, idx, dist, lse2_W, lse2_b, partLse);
  bn_final_kernel<<<1, D2C, 0, stream>>>(partLse, nLseBlk, D2C, lse2_g, lse2_be, cntLse, stLse2);
  pool_kernel<<<(NB * NPTS) / 8, 256, 0, stream>>>(coords, idx, dist, lse2_W, lse2_b, stLse2,
                                                   p1, stP1, 1,
                                                   pool2_sW, pool2_W, pool2_b, DOUTC, p2);
  stats_partial_kernel<<<dim3(DOUTC, 16), 256, 0, stream>>>(p2, DOUTC, partP);
  bn_final_kernel<<<1, DOUTC, 0, stream>>>(partP, 16, DOUTC, pool2_g, pool2_be, cntBN, stP2);

  // final mlp2 + shortcut + leaky 0.01
  final_kernel<<<(NB * NPTS) / 256, 256, 0, stream>>>(p2, stP2, scpre, stSc, mlp2_W, mlp2_b, out);
}